// GaussianSplatting_23046794510522
// MI455X (gfx1250) — compile-verified
//
#include <hip/hip_runtime.h>
#include <hip/hip_bf16.h>
#include <stdint.h>

// Native vector types (usable with __builtin_nontemporal_* and "s"/"v" asm tuples).
typedef float    v4f __attribute__((ext_vector_type(4)));
typedef uint32_t v4u __attribute__((ext_vector_type(4)));
typedef uint32_t v8u __attribute__((ext_vector_type(8)));

#define LDS_SPACE __attribute__((address_space(3)))

// Byte offset of a shared-memory object from the wave's LDS base
// (addrspace(3) pointers are 32-bit LDS offsets on AMDGPU).
__device__ __forceinline__ uint32_t lds_byte_offset(const void* p) {
    return (uint32_t)(uintptr_t)(LDS_SPACE const void*)p;
}

__device__ __forceinline__ void compute_cov(float s0, float s1, float s2,
                                            float r, float x, float y, float z,
                                            float* c /*6: 00,01,02,11,12,22*/) {
    float sx = __expf(s0), sy = __expf(s1), sz = __expf(s2);
    float n2  = r * r + x * x + y * y + z * z;
    float inv = 1.0f / fmaxf(sqrtf(n2), 1e-12f);
    r *= inv; x *= inv; y *= inv; z *= inv;

    float xx = x * x, yy = y * y, zzq = z * z;
    float xy = x * y, xz = x * z, yz = y * z;
    float rx = r * x, ry = r * y, rz = r * z;

    float m00 = (1.0f - 2.0f * (yy + zzq)) * sx;
    float m01 = (2.0f * (xy - rz)) * sy;
    float m02 = (2.0f * (xz + ry)) * sz;
    float m10 = (2.0f * (xy + rz)) * sx;
    float m11 = (1.0f - 2.0f * (xx + zzq)) * sy;
    float m12 = (2.0f * (yz - rx)) * sz;
    float m20 = (2.0f * (xz - ry)) * sx;
    float m21 = (2.0f * (yz + rx)) * sy;
    float m22 = (1.0f - 2.0f * (xx + yy)) * sz;

    c[0] = m00 * m00 + m01 * m01 + m02 * m02;   // c00
    c[1] = m00 * m10 + m01 * m11 + m02 * m12;   // c01
    c[2] = m00 * m20 + m01 * m21 + m02 * m22;   // c02
    c[3] = m10 * m10 + m11 * m11 + m12 * m12;   // c11
    c[4] = m10 * m20 + m11 * m21 + m12 * m22;   // c12
    c[5] = m20 * m20 + m21 * m21 + m22 * m22;   // c22
}

__global__ __launch_bounds__(256) void GaussianSplatting_23046794510522_kernel(
    const float* __restrict__ scaling,   // (N,3)
    const float* __restrict__ rotation,  // (N,4)
    float* __restrict__ cov,             // (N,3,3)
    int n_points)
{
    __shared__ v4f s_scale4[192];  // 768 floats  = 256 pts * 3  (3072 B)
    __shared__ v4f s_cov4[576];    // 2304 floats = 256 pts * 9  (9216 B)

    const int tid       = threadIdx.x;
    const int blockBase = (int)blockIdx.x * 256;

    if (blockBase + 256 <= n_points) {
        // ---------- fast path: full block of 256 points ----------

        // Stage this block's scaling chunk (3072 B, 16B-aligned) into LDS via the
        // CDNA5 async global->LDS engine (ASYNCcnt-tracked), contiguous b128 traffic.
        if (tid < 192) {
            uint64_t ga = (uint64_t)(uintptr_t)scaling
                        + (uint64_t)blockBase * 12u + (uint32_t)tid * 16u;
            uint32_t la = lds_byte_offset(&s_scale4[0]) + (uint32_t)tid * 16u;
            asm volatile("global_load_async_to_lds_b128 %0, %1, off"
                         :: "v"(la), "v"(ga) : "memory");
        }
        asm volatile("s_wait_asynccnt 0x0" ::: "memory");
        __syncthreads();

        // rotation: one aligned float4 per lane, streaming (NT) hint
        // (working set 537 MB ~ 2.8x the 192 MB L2 -> pure streaming).
        const int n  = blockBase + tid;
        v4f q = __builtin_nontemporal_load((const v4f*)rotation + n);

        const float* s_scale = (const float*)s_scale4;  // stride-3 dwords: bank-conflict-free
        float c[6];
        compute_cov(s_scale[tid * 3 + 0], s_scale[tid * 3 + 1], s_scale[tid * 3 + 2],
                    q.x, q.y, q.z, q.w, c);

        // Stage the 9 outputs in LDS (stride-9 dwords: conflict-free on 64 banks).
        float* sc = (float*)s_cov4 + tid * 9;
        sc[0] = c[0]; sc[1] = c[1]; sc[2] = c[2];
        sc[3] = c[1]; sc[4] = c[3]; sc[5] = c[4];
        sc[6] = c[2]; sc[7] = c[4]; sc[8] = c[5];
        __syncthreads();

        // Ship the whole 9216 B block region LDS -> global with ONE Tensor Data
        // Mover op (TENSORcnt-tracked). Issued by wave 0 only; the guard must be
        // a *scalar* branch because TDM ignores EXEC.
        if (__builtin_amdgcn_readfirstlane((int)tid) == 0) {
            uint64_t ga = (uint64_t)(uintptr_t)cov + (uint64_t)blockIdx.x * 9216u;

            // D# group 0: count=1 | lds_addr | global_addr[56:0] | type=2 ("image")
            v4u g0;
            g0[0] = 1u;                                           // count=1, user desc
            g0[1] = lds_byte_offset(&s_cov4[0]);                  // lds_addr (bytes)
            g0[2] = (uint32_t)(ga & 0xFFFFFFFFu);                 // global_addr[31:0]
            g0[3] = (uint32_t)((ga >> 32) & 0x01FFFFFFu)          // global_addr[56:32]
                    | (2u << 30);                                 // type=2

            // D# group 1: 1-row tile of 2304 dwords (data_size=4B).
            v8u g1;
            g1[0] = 0x00020000u;                 // workgroup_mask=0, data_size=2 (4B)
            g1[1] = (2304u & 0xFFFFu) << 16;     // tensor_dim0[15:0]  -> bits 63:48
            g1[2] = (2304u >> 16)                // tensor_dim0[31:16] -> bits 79:64
                    | (1u << 16);                // tensor_dim1[15:0]=1 -> bits 95:80
            g1[3] = (2304u & 0xFFFFu) << 16;     // tile_dim0=2304     -> bits 127:112
            g1[4] = 0u;                          // tile_dim1=0, tile_dim2=0 (unused)
            g1[5] = 2304u;                       // tensor_dim0_stride[31:0]
            g1[6] = 0u;                          // stride0 hi / stride1 lo
            g1[7] = 0u;

            asm volatile("tensor_store_from_lds %0, %1" :: "s"(g0), "s"(g1) : "memory");
            // Drain TENSORcnt (s_endpgm also implies wait-idle, keeping LDS alive).
            asm volatile("s_wait_tensorcnt 0x0" ::: "memory");
        }
    } else {
        // ---------- tail path (unused for N = 2^23, kept for generality) ----------
        const int n = blockBase + tid;
        if (n < n_points) {
            v4f q = __builtin_nontemporal_load((const v4f*)rotation + n);
            float c[6];
            compute_cov(scaling[n * 3 + 0], scaling[n * 3 + 1], scaling[n * 3 + 2],
                        q.x, q.y, q.z, q.w, c);
            float* o = cov + (size_t)n * 9;
            o[0] = c[0]; o[1] = c[1]; o[2] = c[2];
            o[3] = c[1]; o[4] = c[3]; o[5] = c[4];
            o[6] = c[2]; o[7] = c[4]; o[8] = c[5];
        }
    }
}

extern "C" void kernel_launch(void* const* d_in, const int* in_sizes, int n_in,
                              void* d_out, int out_size, void* d_ws, size_t ws_size,
                              hipStream_t stream) {
    (void)n_in; (void)out_size; (void)d_ws; (void)ws_size;
    const float* scaling  = (const float*)d_in[0];   // (N,3) fp32
    const float* rotation = (const float*)d_in[1];   // (N,4) fp32
    float*       cov      = (float*)d_out;           // (N,3,3) fp32

    const int n_points = in_sizes[0] / 3;
    const int blocks   = (n_points + 255) / 256;
    GaussianSplatting_23046794510522_kernel<<<blocks, 256, 0, stream>>>(
        scaling, rotation, cov, n_points);
}